// NgramHead_43250320671331
// MI455X (gfx1250) — compile-verified
//
#include <hip/hip_runtime.h>

typedef __attribute__((ext_vector_type(2))) float v2f;
typedef __attribute__((ext_vector_type(4))) float v4f;
typedef __attribute__((ext_vector_type(8))) float v8f;
typedef __attribute__((ext_vector_type(4))) unsigned int v4u;
typedef __attribute__((ext_vector_type(4))) int v4i;
typedef __attribute__((ext_vector_type(8))) int v8i;

#define BB 4
#define LL 2048
#define DD 1024

#define TM 128
#define TN 128
#define TK 16
#define XPAD 17    // X tile LDS row stride (floats): TDM pad 16DW interval + 1DW
#define WPAD 136   // W tile LDS row stride (floats): TDM pad 128DW interval + 8DW

// ---------------------------------------------------------------------------
// Kernel 1: pack trigram keys. key[b,i] = ids[i-2]<<10 | ids[i-1]<<5 | ids[i]
// for i>=2; zero-padded trigram (key==0) for i<2, matching the reference pad.
// ---------------------------------------------------------------------------
__global__ void ngram_keys_kernel(const int* __restrict__ ids,
                                  int* __restrict__ keys) {
    int idx = blockIdx.x * blockDim.x + threadIdx.x;   // over B*L
    if (idx >= BB * LL) return;
    int i = idx & (LL - 1);
    int key = 0;
    if (i >= 2) {
        int a = ids[idx - 2] & 31;
        int b = ids[idx - 1] & 31;
        int c = ids[idx]     & 31;
        key = (a << 10) | (b << 5) | c;
    }
    keys[idx] = key;
}

// ---------------------------------------------------------------------------
// Kernel 2: sparse causal mean-pool. One block per (b,i): scan j<i for key
// matches (keys sit in L2), gather match indices into LDS, then each thread
// mean-reduces a 4-float column slice over the (tiny) match list.
// ---------------------------------------------------------------------------
__global__ __launch_bounds__(256)
void ngram_pool_kernel(const float* __restrict__ hidden,
                       const int* __restrict__ keys,
                       float* __restrict__ pooled) {
    __shared__ int s_list[LL];
    __shared__ int s_cnt;
    const int bi  = blockIdx.x;          // 0..B*L-1
    const int i   = bi & (LL - 1);
    const int b   = bi >> 11;            // L = 2048
    const int tid = threadIdx.x;         // 256 threads

    if (tid == 0) s_cnt = 0;
    __syncthreads();

    const int* krow = keys + b * LL;
    const int mykey = krow[i];
    for (int j = tid; j < i; j += 256) {
        if (krow[j] == mykey) {
            int p = atomicAdd(&s_cnt, 1);
            s_list[p] = j;
        }
    }
    __syncthreads();

    const int n = s_cnt;
    v4f acc = {0.0f, 0.0f, 0.0f, 0.0f};
    const float* hb = hidden + (size_t)b * LL * DD;
    for (int t = 0; t < n; ++t) {
        const float* hr = hb + (size_t)s_list[t] * DD + tid * 4;
        v4f h = *(const v4f*)hr;
        acc += h;
    }
    const float inv = (n > 0) ? (1.0f / (float)n) : 0.0f;
    acc *= inv;
    *(v4f*)(pooled + (size_t)bi * DD + tid * 4) = acc;
}

// ---------------------------------------------------------------------------
// TDM issue helper: one 2D tile (tile0 contiguous elems x tile1 rows of f32,
// row stride stride0 elems) from global -> LDS, with LDS row padding.
// D# bitfields per CDNA5 ISA 8.3/8.4:
//   G0: [1:0]=count=1 ... [63:32]=lds_addr, [120:64]=global_addr, [127:126]=2
//   G1: [17:16]=data_size(2=4B), [20]=pad_enable, [24:22]=pad_interval,
//       [31:25]=pad_amount, [79:48]=tensor_dim0, [111:80]=tensor_dim1,
//       [127:112]=tile_dim0, [143:128]=tile_dim1, [207:160]=dim0_stride
// This toolchain exposes the 6-arg builtin:
//   (v4u g0, v8i g1, v4i g2, v4i g3, v8i g4_unused, i32 cpol)
// ---------------------------------------------------------------------------
__device__ __forceinline__ void tdm_load_2d(unsigned int lds_addr,
                                            const void* gptr,
                                            unsigned int tile0,
                                            unsigned int tile1,
                                            unsigned long long stride0,
                                            unsigned int pad_interval,
                                            unsigned int pad_amount) {
    const unsigned long long ga = (unsigned long long)(uintptr_t)gptr;
    v4u g0;
    g0[0] = 1u;                                   // count=1, user descriptor
    g0[1] = lds_addr;                             // LDS byte address
    g0[2] = (unsigned int)(ga & 0xFFFFFFFFull);   // global_addr[31:0]
    g0[3] = (unsigned int)((ga >> 32) & 0x01FFFFFFull) | (2u << 30); // type=2

    const unsigned int td0 = tile0;               // tensor dims = tile dims
    const unsigned int td1 = tile1;               // (tile fully in-bounds)
    v8i g1;
    g1[0] = (int)((2u << 16)                      // data_size = 4 bytes
                | (1u << 20)                      // pad_enable
                | (pad_interval << 22)
                | (pad_amount << 25));
    g1[1] = (int)((td0 & 0xFFFFu) << 16);         // tensor_dim0[15:0] @bit48
    g1[2] = (int)(((td0 >> 16) & 0xFFFFu) | ((td1 & 0xFFFFu) << 16));
    g1[3] = (int)(((td1 >> 16) & 0xFFFFu) | (tile0 << 16));  // tile_dim0
    g1[4] = (int)(tile1 & 0xFFFFu);               // tile_dim1 (tile_dim2=0)
    g1[5] = (int)(unsigned int)(stride0 & 0xFFFFFFFFull);    // dim0_stride lo
    g1[6] = (int)(unsigned int)((stride0 >> 32) & 0xFFFFull);// dim0_stride hi
    g1[7] = 0;

    v4i g2 = {0, 0, 0, 0};
    v4i g3 = {0, 0, 0, 0};
    v8i g4 = {0, 0, 0, 0, 0, 0, 0, 0};            // unused VADDR4 slot
    __builtin_amdgcn_tensor_load_to_lds(g0, g1, g2, g3, g4, 0);
}

// ---------------------------------------------------------------------------
// Kernel 3: fused dual GEMM with f32 WMMA + TDM double-buffered staging.
//   out = X@W1 + P@W2 + (b1+b2)
// 128x128 tile / block, 8 waves, each wave = 2x4 fragments of 16x16.
// Each wave TDM-loads its own slice of the next K-block (16 X rows, 2 W rows)
// while the current one feeds V_WMMA_F32_16X16X4_F32.
// ---------------------------------------------------------------------------
__global__ __launch_bounds__(256)
void ngram_out_kernel(const float* __restrict__ X,    // hidden [M,K]
                      const float* __restrict__ P,    // pooled [M,K]
                      const float* __restrict__ W1,   // [K,N]
                      const float* __restrict__ W2,   // [K,N]
                      const float* __restrict__ b1,   // [N]
                      const float* __restrict__ b2,   // [N]
                      float* __restrict__ out) {      // [M,N]
    __shared__ float Xs[2][TM * XPAD];   // 2 x 8704 B
    __shared__ float Ws[2][TK * WPAD];   // 2 x 8704 B

    const int tid  = threadIdx.x;
    const int lane = tid & 31;
    const int wave = __builtin_amdgcn_readfirstlane(tid >> 5);  // uniform
    const int wm   = wave & 3;        // m offset = wm*32
    const int wn   = wave >> 2;       // n offset = wn*64
    const int half = lane >> 4;       // 0: lanes 0-15, 1: lanes 16-31
    const int r    = lane & 15;

    const int tileM = blockIdx.x * TM;
    const int tileN = blockIdx.y * TN;

    // LDS byte addresses of this wave's TDM destination slices (generic
    // shared pointer low 32 bits == LDS address per flat-aperture mapping).
    const unsigned int xs_lds[2] = {
        (unsigned int)(uintptr_t)&Xs[0][wave * 16 * XPAD],
        (unsigned int)(uintptr_t)&Xs[1][wave * 16 * XPAD]};
    const unsigned int ws_lds[2] = {
        (unsigned int)(uintptr_t)&Ws[0][wave * 2 * WPAD],
        (unsigned int)(uintptr_t)&Ws[1][wave * 2 * WPAD]};

    v8f acc[2][4];
    // Bias init: within a fragment every lane owns one column (= r), all 8
    // row-VGPRs share that column -> broadcast b1[n]+b2[n] into all 8 slots.
    #pragma unroll
    for (int sn = 0; sn < 4; ++sn) {
        const int n = tileN + wn * 64 + sn * 16 + r;
        const float bs = b1[n] + b2[n];
        v8f c;
        #pragma unroll
        for (int v = 0; v < 8; ++v) c[v] = bs;
        acc[0][sn] = c;
        acc[1][sn] = c;
    }

    const int NT = 2 * (DD / TK);   // 2 passes x 64 K-blocks = 128 tiles

    // Issue one tile pair: this wave's slices of X/P block and W1/W2 block.
    auto issue = [&](int t) {
        const int    buf = t & 1;
        const int    pass = t >> 6;
        const int    k0  = (t & 63) * TK;
        const float* A = pass ? P  : X;
        const float* W = pass ? W2 : W1;
        // 16 rows x 16 cols of A, row stride K; LDS stride 17 DW (pad 16+1)
        tdm_load_2d(xs_lds[buf],
                    A + (size_t)(tileM + wave * 16) * DD + k0,
                    /*tile0=*/16, /*tile1=*/16, /*stride0=*/DD,
                    /*pad_interval=*/3 /*16 DW*/, /*pad_amount=*/0 /*1 DW*/);
        // 2 rows x 128 cols of W, row stride N; LDS stride 136 DW (pad 128+8)
        tdm_load_2d(ws_lds[buf],
                    W + (size_t)(k0 + wave * 2) * DD + tileN,
                    /*tile0=*/128, /*tile1=*/2, /*stride0=*/DD,
                    /*pad_interval=*/6 /*128 DW*/, /*pad_amount=*/7 /*8 DW*/);
    };

    issue(0);   // prologue

    #pragma unroll 1
    for (int t = 0; t < NT; ++t) {
        // Overlap: issue tile t+1 into the other buffer, then wait only for
        // tile t's two (in-order) TDM ops.
        if (t + 1 < NT) {
            issue(t + 1);
            __builtin_amdgcn_s_wait_tensorcnt(2);
        } else {
            __builtin_amdgcn_s_wait_tensorcnt(0);
        }
        __syncthreads();   // publish this wave's slices to all waves

        const float* XsB = &Xs[t & 1][0];
        const float* WsB = &Ws[t & 1][0];

        #pragma unroll
        for (int kk = 0; kk < TK; kk += 4) {
            // A fragment (16x4): half 0 -> K={kk,kk+1}, half 1 -> K={kk+2,kk+3}
            v2f afr[2];
            #pragma unroll
            for (int sm = 0; sm < 2; ++sm) {
                const float* p = &XsB[(wm * 32 + sm * 16 + r) * XPAD + kk + half * 2];
                afr[sm][0] = p[0];
                afr[sm][1] = p[1];
            }
            // B fragment (4x16): VGPR0 rows {kk,kk+2}, VGPR1 rows {kk+1,kk+3}
            v2f bfr[4];
            #pragma unroll
            for (int sn = 0; sn < 4; ++sn) {
                const int col = wn * 64 + sn * 16 + r;
                bfr[sn][0] = WsB[(kk + half * 2 + 0) * WPAD + col];
                bfr[sn][1] = WsB[(kk + half * 2 + 1) * WPAD + col];
            }
            #pragma unroll
            for (int sm = 0; sm < 2; ++sm) {
                #pragma unroll
                for (int sn = 0; sn < 4; ++sn) {
                    acc[sm][sn] = __builtin_amdgcn_wmma_f32_16x16x4_f32(
                        /*neg_a=*/false, afr[sm],
                        /*neg_b=*/false, bfr[sn],
                        /*c_mod=*/(short)0, acc[sm][sn],
                        /*reuse_a=*/false, /*reuse_b=*/false);
                }
            }
        }
        __syncthreads();   // all waves done reading buf before it is re-DMAed
    }

    // Store: VGPR v holds row (v + half*8) of the 16x16 fragment, column r.
    #pragma unroll
    for (int sm = 0; sm < 2; ++sm) {
        const int mbase = tileM + wm * 32 + sm * 16 + half * 8;
        #pragma unroll
        for (int sn = 0; sn < 4; ++sn) {
            const int col = tileN + wn * 64 + sn * 16 + r;
            #pragma unroll
            for (int v = 0; v < 8; ++v) {
                out[(size_t)(mbase + v) * DD + col] = acc[sm][sn][v];
            }
        }
    }
}

// ---------------------------------------------------------------------------
extern "C" void kernel_launch(void* const* d_in, const int* in_sizes, int n_in,
                              void* d_out, int out_size, void* d_ws, size_t ws_size,
                              hipStream_t stream) {
    const float* hidden = (const float*)d_in[0];
    const int*   ids    = (const int*)d_in[1];
    const float* W1     = (const float*)d_in[2];
    const float* b1     = (const float*)d_in[3];
    const float* W2     = (const float*)d_in[4];
    const float* b2     = (const float*)d_in[5];
    float*       out    = (float*)d_out;

    int*   keys   = (int*)d_ws;                            // B*L ints (32 KB)
    float* pooled = (float*)((char*)d_ws + 64 * 1024);     // B*L*D floats (32 MB)

    ngram_keys_kernel<<<(BB * LL + 255) / 256, 256, 0, stream>>>(ids, keys);
    ngram_pool_kernel<<<BB * LL, 256, 0, stream>>>(hidden, keys, pooled);

    dim3 grid(BB * LL / TM, DD / TN);
    ngram_out_kernel<<<grid, 256, 0, stream>>>(hidden, pooled, W1, W2, b1, b2, out);
}